// Attention_54451595378952
// MI455X (gfx1250) — compile-verified
//
#include <hip/hip_runtime.h>
#include <math.h>

typedef __attribute__((ext_vector_type(2))) float v2f;
typedef __attribute__((ext_vector_type(8))) float v8f;

#define BATCH 8
#define SEQ   3136
#define CIN   147
#define DMODEL 64
#define NQKV  192           // 3*64
#define NTILES_QKV 12       // 192/16
#define QTILES 196          // 3136/16
#define KVTILES 196
#define SCALE 0.125f        // 64^-0.5
#define LSTR  68            // LDS row stride (floats): conflict-free rotation

// wmma f32 16x16x4: D = A(16x4) * B(4x16) + C
__device__ __forceinline__ v8f wmma_f32(v2f a, v2f b, v8f c) {
    return __builtin_amdgcn_wmma_f32_16x16x4_f32(false, a, false, b, (short)0, c,
                                                 false, false);
}

// XOR-butterfly inside 16-lane halves via ds_swizzle (group-of-32 mode:
// imm = {0, xor[4:0], or[4:0]=0, and[4:0]=0x1f}); masks < 16 never cross halves.
#define SWIZ_XOR_F(v, m) \
    __int_as_float(__builtin_amdgcn_ds_swizzle(__float_as_int(v), (((m) << 10) | 0x1f)))

__device__ __forceinline__ float half_reduce_max(float v) {
    v = fmaxf(v, SWIZ_XOR_F(v, 1));
    v = fmaxf(v, SWIZ_XOR_F(v, 2));
    v = fmaxf(v, SWIZ_XOR_F(v, 4));
    v = fmaxf(v, SWIZ_XOR_F(v, 8));
    return v;
}
__device__ __forceinline__ float half_reduce_sum(float v) {
    v += SWIZ_XOR_F(v, 1);
    v += SWIZ_XOR_F(v, 2);
    v += SWIZ_XOR_F(v, 4);
    v += SWIZ_XOR_F(v, 8);
    return v;
}

// CDNA5 async global->LDS DMA (ASYNCcnt-tracked), GV addressing mode.
__device__ __forceinline__ void async_ld_b128(unsigned lds_byte_off,
                                              const float* gsrc) {
    asm volatile("global_load_async_to_lds_b128 %0, %1, off"
                 :: "v"(lds_byte_off), "v"(gsrc)
                 : "memory");
}
__device__ __forceinline__ void wait_async0() {
    asm volatile("s_wait_asynccnt 0x0" ::: "memory");
}

// ---------------------------------------------------------------------------
// Kernel 1: qkv = x @ W_qkv (25088x147 * 147x192) -> q/k/v scratch.
// One wave per 16-row M tile; branch-free K loop (36 full steps + 1 padded).
// ---------------------------------------------------------------------------
__global__ __launch_bounds__(128) void qkv_kernel(
    const float* __restrict__ x, const float* __restrict__ Wqkv,
    float* __restrict__ q, float* __restrict__ k, float* __restrict__ v) {
    const int lane = threadIdx.x & 31;
    const int wave = threadIdx.x >> 5;
    const int mtile = blockIdx.x * 4 + wave;            // 0..1567
    const int l16  = lane & 15;
    const int half = lane >> 4;
    const float* xr = x + ((size_t)mtile * 16 + l16) * CIN;  // A row (lane = M)

    for (int nt = 0; nt < NTILES_QKV; ++nt) {
        v8f acc = {};
        const int col = nt * 16 + l16;                  // B col (lane = N)
        const float* wc = Wqkv + col;
        for (int ks = 0; ks < 36; ++ks) {               // k0 max = 140+2+1 = 143
            const int k0 = ks * 4 + 2 * half;
            v2f a, b;
            a.x = xr[k0];
            a.y = xr[k0 + 1];
            b.x = wc[(size_t)k0 * NQKV];
            b.y = wc[(size_t)(k0 + 1) * NQKV];
            acc = wmma_f32(a, b, acc);
        }
        {   // tail: k = 144..146, zero-pad k = 147 (branch-free select)
            const int k0 = 144 + 2 * half;              // 144 or 146, in-bounds
            v2f a, b;
            a.x = xr[k0];
            const float ay = xr[145 + half];            // 145 or 146, in-bounds
            a.y = half ? 0.0f : ay;
            b.x = wc[(size_t)k0 * NQKV];
            const float by = wc[(size_t)(145 + half) * NQKV];
            b.y = half ? 0.0f : by;
            acc = wmma_f32(a, b, acc);
        }
        // C/D layout: vgpr i -> row i (+8 for upper half); col = l16
        float* dst = (col < 64) ? q : ((col < 128) ? k : v);
        const int cc = col & 63;
        #pragma unroll
        for (int i = 0; i < 8; ++i) {
            const size_t r = (size_t)mtile * 16 + i + 8 * half;
            dst[r * DMODEL + cc] = acc[i];
        }
    }
}

// ---------------------------------------------------------------------------
// Kernel 2: flash attention + output projection + residual.
// 4 waves / block share a batch; K/V tiles staged once per block into LDS
// with async global->LDS b128 DMA. One wave per 16-row query tile.
// ---------------------------------------------------------------------------
__global__ __launch_bounds__(128) void attn_kernel(
    const float* __restrict__ qg, const float* __restrict__ kg,
    const float* __restrict__ vg, const float* __restrict__ Wproj,
    const float* __restrict__ bproj, float* __restrict__ out) {
    __shared__ float Klds[16 * LSTR];                   // 4.35 KB
    __shared__ float Vlds[16 * LSTR];                   // 4.35 KB
    __shared__ float Pslab[4][16 * LSTR];               // 17.4 KB (per-wave)

    const int tid  = threadIdx.x;
    const int lane = tid & 31;
    const int wave = tid >> 5;
    const int w    = blockIdx.x * 4 + wave;             // 196%4==0 -> same batch
    const int batch = w / QTILES;
    const int qtile = w % QTILES;
    const int l16  = lane & 15;
    const int half = lane >> 4;

    const float* Q = qg + (size_t)batch * SEQ * DMODEL;
    const float* K = kg + (size_t)batch * SEQ * DMODEL;
    const float* V = vg + (size_t)batch * SEQ * DMODEL;
    float* slab = Pslab[wave];

    // block-wide staging map: tile is 1024 contiguous floats; thread t moves
    // chunks t and t+128 (16B each), row = chunk/16, col = (chunk%16)*4
    const int srow0 = tid >> 4;                         // 0..7
    const int scol  = (tid & 15) * 4;                   // 0..60
    const unsigned kl0 = (unsigned)(size_t)&Klds[srow0 * LSTR + scol];
    const unsigned kl1 = (unsigned)(size_t)&Klds[(srow0 + 8) * LSTR + scol];
    const unsigned vl0 = (unsigned)(size_t)&Vlds[srow0 * LSTR + scol];
    const unsigned vl1 = (unsigned)(size_t)&Vlds[(srow0 + 8) * LSTR + scol];

    // Q tile as 16 A-fragments (K = 64), pre-scaled; 8B-aligned b64 loads.
    v2f qa[16];
    {
        const v2f* Qr = (const v2f*)(Q + ((size_t)qtile * 16 + l16) * DMODEL);
        #pragma unroll
        for (int j = 0; j < 16; ++j) {
            v2f t = Qr[2 * j + half];
            t.x *= SCALE; t.y *= SCALE;
            qa[j] = t;
        }
    }

    float mstate[8], lstate[8];
    v8f accO[4] = {};
    #pragma unroll
    for (int i = 0; i < 8; ++i) { mstate[i] = -3.0e38f; lstate[i] = 0.0f; }

    for (int kt = 0; kt < KVTILES; ++kt) {
        const float* Ktile = K + (size_t)kt * 16 * DMODEL;
        const float* Vtile = V + (size_t)kt * 16 * DMODEL;
        async_ld_b128(kl0, Ktile + tid * 4);
        async_ld_b128(kl1, Ktile + tid * 4 + 512);
        async_ld_b128(vl0, Vtile + tid * 4);
        async_ld_b128(vl1, Vtile + tid * 4 + 512);
        if (kt + 1 < KVTILES)                           // global_prefetch_b8
            __builtin_prefetch(K + (size_t)(kt + 1) * 16 * DMODEL, 0, 1);
        wait_async0();
        __syncthreads();

        // S = (Q*scale) @ K^T : 16 WMMAs, B-frags from LDS (b64, bank-clean)
        v8f s = {};
        const v2f* Krow = (const v2f*)(Klds + l16 * LSTR);
        #pragma unroll
        for (int j = 0; j < 16; ++j) s = wmma_f32(qa[j], Krow[2 * j + half], s);

        // online softmax: each row lives across one 16-lane half (C/D layout)
        float alpha[8];
        #pragma unroll
        for (int i = 0; i < 8; ++i) {
            const float rmax = half_reduce_max(s[i]);
            const float nm = fmaxf(mstate[i], rmax);
            alpha[i] = __expf(mstate[i] - nm);
            mstate[i] = nm;
            const float p = __expf(s[i] - nm);
            s[i] = p;
            lstate[i] = lstate[i] * alpha[i] + half_reduce_sum(p);
        }
        #pragma unroll
        for (int nt = 0; nt < 4; ++nt)
            #pragma unroll
            for (int i = 0; i < 8; ++i) accO[nt][i] *= alpha[i];

        // C/D -> A layout transpose of P through this wave's LDS slab
        #pragma unroll
        for (int i = 0; i < 8; ++i) slab[(i + 8 * half) * LSTR + l16] = s[i];
        __syncthreads();
        v2f pa[4];
        {
            const v2f* Prow = (const v2f*)(slab + l16 * LSTR);
            #pragma unroll
            for (int j = 0; j < 4; ++j) pa[j] = Prow[2 * j + half];
        }

        // O += P @ Vtile : 16 WMMAs, V B-frags from LDS
        #pragma unroll
        for (int nt = 0; nt < 4; ++nt)
            #pragma unroll
            for (int j = 0; j < 4; ++j) {
                const int kr = 4 * j + 2 * half;
                v2f vb;
                vb.x = Vlds[kr * LSTR + nt * 16 + l16];
                vb.y = Vlds[(kr + 1) * LSTR + nt * 16 + l16];
                accO[nt] = wmma_f32(pa[j], vb, accO[nt]);
            }
        __syncthreads();   // guards K/V/P LDS reuse next iteration
    }

    // normalize, bounce O through LDS for the 64x64 projection GEMM
    #pragma unroll
    for (int nt = 0; nt < 4; ++nt)
        #pragma unroll
        for (int i = 0; i < 8; ++i)
            slab[(i + 8 * half) * LSTR + nt * 16 + l16] = accO[nt][i] / lstate[i];
    __syncthreads();

    v2f oa[16];
    {
        const v2f* Orow = (const v2f*)(slab + l16 * LSTR);
        #pragma unroll
        for (int j = 0; j < 16; ++j) oa[j] = Orow[2 * j + half];
    }

    // out = O @ W_proj + b_proj + v
    #pragma unroll
    for (int nt = 0; nt < 4; ++nt) {
        v8f accp = {};
        #pragma unroll
        for (int j = 0; j < 16; ++j) {
            const int kr = 4 * j + 2 * half;
            v2f wb;
            wb.x = Wproj[(size_t)kr * DMODEL + nt * 16 + l16];
            wb.y = Wproj[(size_t)(kr + 1) * DMODEL + nt * 16 + l16];
            accp = wmma_f32(oa[j], wb, accp);
        }
        const int c = nt * 16 + l16;
        #pragma unroll
        for (int i = 0; i < 8; ++i) {
            const size_t r = (size_t)qtile * 16 + i + 8 * half;
            const size_t idx = ((size_t)batch * SEQ + r) * DMODEL + c;
            out[idx] = accp[i] + bproj[c] + V[r * DMODEL + c];
        }
    }
}

extern "C" void kernel_launch(void* const* d_in, const int* in_sizes, int n_in,
                              void* d_out, int out_size, void* d_ws, size_t ws_size,
                              hipStream_t stream) {
    const float* x     = (const float*)d_in[0];  // (8, 3136, 147)
    const float* Wqkv  = (const float*)d_in[1];  // (147, 192)
    const float* Wproj = (const float*)d_in[2];  // (64, 64)
    const float* bproj = (const float*)d_in[3];  // (64,)
    float* out = (float*)d_out;                  // (8, 3136, 64)

    const size_t per_mat = (size_t)BATCH * SEQ * DMODEL;   // 1,605,632 floats
    float* q = (float*)d_ws;
    float* k = q + per_mat;
    float* v = k + per_mat;

    qkv_kernel<<<392, 128, 0, stream>>>(x, Wqkv, q, k, v);
    attn_kernel<<<392, 128, 0, stream>>>(q, k, v, Wproj, bproj, out);
}